// HGT_661424964321
// MI455X (gfx1250) — compile-verified
//
#include <hip/hip_runtime.h>
#include <cstdint>

// ---------------------------------------------------------------------------
// HGT forward for MI455X (gfx1250, wave32, WMMA).
//
//  * a_rel/m_rel folded into K/V projection weights -> all dense ops are
//    (N,128)@(128,128) GEMMs on V_WMMA_F32_16X16X4_F32 (full f32 precision).
//  * All GEMM weights are pre-packed into WMMA B-fragment order so the inner
//    loop is 1 A b64 + 8 lane-coalesced B b64 loads per K-step, no EXEC games
//    (out-of-range rows are clamped, never stored).
//  * Edge phase: ordered-uint atomicMax segment max, then fused
//    den += ex, agg += ex*w*msg pass; node-level divide + tanh-gelu; output
//    GEMM with fused sigmoid skip-gate.
//
// d_in layout assumption (params = JAX pytree, nested dict keys SORTED):
//   0..2   x_patient x_disease x_drug
//   3..10  ei_phd,w_phd, ei_dhp,w_dhp, ei_gtd,w_gtd, ei_dtg,w_dtg (ei is 2xE)
//   11..16 in: (disease w,b)(drug w,b)(patient w,b)
//   layer l base L=17+39l, groups a(+0) k(+6) q(+12) rel(+18, sorted
//   dhp,dtg,gtd,phd each a_rel,m_rel,p_rel) skip(+30) v(+33)
// ---------------------------------------------------------------------------

typedef float v2f __attribute__((ext_vector_type(2)));
typedef float v8f __attribute__((ext_vector_type(8)));

#define HID 128

// packed B-fragment index (in v2f units) for weight element (r=K idx, col)
__device__ __forceinline__ size_t pack_idx(int r, int col) {
  int ks = r >> 2, hi = (r >> 1) & 1;
  int j = col >> 4, lane = hi * 16 + (col & 15);
  return ((size_t)ks * 8 + j) * 32 + lane;          // *2 (+ r&1) for float idx
}

// ---- pack a KxHID row-major weight into WMMA B-fragment order --------------
__global__ __launch_bounds__(256) void hgt_pack_w(
    const float* __restrict__ W, float* __restrict__ PW, int K)
{
  int idx = blockIdx.x * blockDim.x + threadIdx.x;
  if (idx >= K * HID) return;
  int r = idx >> 7, col = idx & 127;
  PW[pack_idx(r, col) * 2 + (r & 1)] = W[idx];
}

// ---- GEMM: Y = epilogue(X(NxK) @ W(Kx128) + bias), W pre-packed ------------
// mode 0: plain   mode 1: relu   mode 2: out = beta*(..) + (1-beta)*Xin
__global__ __launch_bounds__(256) void hgt_gemm_wmma(
    const float* __restrict__ X, const float* __restrict__ PW,
    const float* __restrict__ bias, float* __restrict__ Y,
    const float* __restrict__ Xin, const float* __restrict__ skip,
    int N, int K, int mode)
{
  const int lane = threadIdx.x & 31;
  const int wave = threadIdx.x >> 5;
  const int row0 = (blockIdx.x * 8 + wave) * 16;    // 16 rows per wave
  if (row0 >= N) return;
  const int m  = lane & 15;
  const int hi = lane >> 4;                         // 0: K{0,1}, 1: K{2,3}
  const int row = min(row0 + m, N - 1);             // clamp: invalid rows never stored
  const v2f* xr = (const v2f*)(X + (size_t)row * K);
  const v2f* wp = (const v2f*)PW + lane;

  v8f acc[8] = {};
  const int steps = K >> 2;
#pragma unroll 2
  for (int ks = 0; ks < steps; ++ks) {
    const v2f a = xr[ks * 2 + hi];                  // A frag: one b64
    const v2f* wk = wp + (size_t)ks * 256;
#pragma unroll
    for (int j = 0; j < 8; ++j) {                   // 8 col tiles -> 128 cols
      const v2f b = wk[j * 32];                     // lane-coalesced b64
      acc[j] = __builtin_amdgcn_wmma_f32_16x16x4_f32(
          false, a, false, b, (short)0, acc[j], false, false);
    }
  }
  float beta = 0.0f;
  if (mode == 2) beta = 1.0f / (1.0f + expf(-skip[0]));
#pragma unroll
  for (int j = 0; j < 8; ++j) {
    const int col = j * 16 + m;
    const float bv = bias[col];
#pragma unroll
    for (int i = 0; i < 8; ++i) {
      const int r = row0 + i + 8 * hi;              // C/D layout: M = i + 8*hi
      if (r < N) {
        float v = acc[j][i] + bv;
        if (mode == 1) v = fmaxf(v, 0.0f);
        if (mode == 2) v = beta * v + (1.0f - beta) * Xin[(size_t)r * HID + col];
        Y[(size_t)r * HID + col] = v;
      }
    }
  }
}

// ---- fold a_rel/m_rel (8x16x16) into packed 128x128 weight + bias ----------
__global__ __launch_bounds__(256) void hgt_fold_rel(
    const float* __restrict__ Wsrc, const float* __restrict__ bsrc,
    const float* __restrict__ rel,  // (h,d,f) = h*256 + d*16 + f
    float* __restrict__ PWdst, float* __restrict__ bdst)
{
  int idx = blockIdx.x * blockDim.x + threadIdx.x;
  if (idx < 128 * 128) {
    int r = idx >> 7, col = idx & 127;
    int h = col >> 4, f = col & 15;
    const float* a = rel + h * 256 + f;             // stride 16 over d
    const float* w = Wsrc + r * 128 + h * 16;
    float s = 0.0f;
#pragma unroll
    for (int d = 0; d < 16; ++d) s += w[d] * a[d * 16];
    PWdst[pack_idx(r, col) * 2 + (r & 1)] = s;      // write packed directly
  } else if (idx < 128 * 128 + 128) {
    int col = idx - 128 * 128;
    int h = col >> 4, f = col & 15;
    float s = 0.0f;
#pragma unroll
    for (int d = 0; d < 16; ++d) s += bsrc[h * 16 + d] * rel[h * 256 + d * 16 + f];
    bdst[col] = s;
  }
}

__device__ __forceinline__ unsigned f2ord(float f) {
  unsigned u = __float_as_uint(f);
  return (u & 0x80000000u) ? ~u : (u | 0x80000000u);
}
__device__ __forceinline__ float ord2f(unsigned u) {
  return __uint_as_float((u & 0x80000000u) ? (u & 0x7FFFFFFFu) : ~u);
}

// ---- pass 1: scores + running max ------------------------------------------
__global__ __launch_bounds__(256) void hgt_edge_score(
    const int* __restrict__ src, const int* __restrict__ dst,
    const float* __restrict__ q,   // dst-type base already applied
    const float* __restrict__ kr,  // relation base already applied
    const float* __restrict__ prel,
    float* __restrict__ score, unsigned* __restrict__ mmax, int E)
{
  int t = blockIdx.x * blockDim.x + threadIdx.x;
  if (t >= E * 8) return;
  int e = t >> 3, h = t & 7;
  int s = src[e], d = dst[e];
  const float4* qa = (const float4*)(q  + (size_t)d * HID + h * 16);
  const float4* ka = (const float4*)(kr + (size_t)s * HID + h * 16);
  float acc = 0.0f;
#pragma unroll
  for (int i = 0; i < 4; ++i) {
    float4 qv = qa[i], kv = ka[i];
    acc += qv.x * kv.x + qv.y * kv.y + qv.z * kv.z + qv.w * kv.w;
  }
  acc *= prel[h] * 0.25f;                           // 1/sqrt(D), D=16
  score[t] = acc;
  atomicMax(mmax + (size_t)d * 8 + h, f2ord(acc));
}

// ---- pass 2: ex, denominator, weighted message aggregation -----------------
__global__ __launch_bounds__(256) void hgt_edge_accum(
    const int* __restrict__ src, const int* __restrict__ dst,
    const float* __restrict__ ew, const float* __restrict__ score,
    const float* __restrict__ mg, const unsigned* __restrict__ mmax,
    float* __restrict__ den, float* __restrict__ agg, int E)
{
  int t = blockIdx.x * blockDim.x + threadIdx.x;
  if (t >= E * 8) return;
  int e = t >> 3, h = t & 7;
  int s = src[e], d = dst[e];
  float m = ord2f(mmax[(size_t)d * 8 + h]);
  float ex = expf(score[t] - m);
  atomicAdd(den + (size_t)d * 8 + h, ex);
  float c = ex * ew[e];
  const float4* mv = (const float4*)(mg + (size_t)s * HID + h * 16);
  float* ag = agg + (size_t)d * HID + h * 16;
#pragma unroll
  for (int i = 0; i < 4; ++i) {
    float4 v = mv[i];
    atomicAdd(ag + i * 4 + 0, c * v.x);
    atomicAdd(ag + i * 4 + 1, c * v.y);
    atomicAdd(ag + i * 4 + 2, c * v.z);
    atomicAdd(ag + i * 4 + 3, c * v.w);
  }
}

// ---- normalize by denominator + tanh-gelu (in place) -----------------------
__global__ __launch_bounds__(256) void hgt_norm_gelu(
    float* __restrict__ agg, const float* __restrict__ den, int Nrows)
{
  int t = blockIdx.x * blockDim.x + threadIdx.x;
  if (t >= Nrows * HID) return;
  int n = t >> 7, c = t & 127, h = c >> 4;
  float dd = den[(size_t)n * 8 + h];
  float v = agg[t] / fmaxf(dd, 1e-16f);
  float u = 0.7978845608028654f * (v + 0.044715f * v * v * v);
  agg[t] = 0.5f * v * (1.0f + tanhf(u));
}

// ---------------------------------------------------------------------------
extern "C" void kernel_launch(void* const* d_in, const int* in_sizes, int n_in,
                              void* d_out, int out_size, void* d_ws, size_t ws_size,
                              hipStream_t stream)
{
  (void)in_sizes; (void)n_in; (void)out_size; (void)ws_size;
  const int NT = 200000, NE = 250000;
  const int    Nt[3]   = {100000, 50000, 50000};    // patient, disease, drug
  const int    Kin[3]  = {64, 128, 64};
  const size_t base[3] = {0, 100000, 150000};
  const float* x_in[3] = {(const float*)d_in[0], (const float*)d_in[1], (const float*)d_in[2]};

  const int* ei[4]; const float* ew[4];
  for (int r = 0; r < 4; ++r) { ei[r] = (const int*)d_in[3 + 2*r]; ew[r] = (const float*)d_in[4 + 2*r]; }
  const int    rsrc[4]    = {0, 1, 2, 1};           // ETS order phd,dhp,gtd,dtg
  const int    rdst[4]    = {1, 0, 1, 2};
  const size_t relbase[4] = {0, 100000, 150000, 200000};

  const int pslot[3] = {2, 0, 1};                   // type -> sorted slot (d<dr<p)
  const int rslot[4] = {3, 0, 2, 1};                // rel  -> sorted slot
  auto P = [&](int idx) { return (const float*)d_in[idx]; };

  // workspace layout (floats); total ~179M floats (~715 MB)
  float* ws = (float*)d_ws;
  size_t o = 0;
  float* bufA   = ws + o; o += (size_t)NT * HID;
  float* bufB   = ws + o; o += (size_t)NT * HID;
  float* qbuf   = ws + o; o += (size_t)NT * HID;
  float* krbuf  = ws + o; o += (size_t)250000 * HID;
  float* mgbuf  = ws + o; o += (size_t)250000 * HID;
  float* aggbuf = ws + o; o += (size_t)NT * HID;
  unsigned* mmax = (unsigned*)(ws + o); o += (size_t)NT * 8;
  float* den    = ws + o; o += (size_t)NT * 8;
  float* scoreb = ws + o; o += (size_t)4 * NE * 8;
  float* foldW  = ws + o; o += (size_t)8 * 128 * 128;   // packed (per relation k,v)
  float* foldB  = ws + o; o += (size_t)8 * 128;
  float* packIn = ws + o; o += (size_t)3 * 128 * 128;   // packed in-proj weights
  float* packQ  = ws + o; o += (size_t)3 * 128 * 128;   // packed q weights (per layer)
  float* packA  = ws + o; o += (size_t)3 * 128 * 128;   // packed a weights (per layer)

  auto gemm = [&](const float* X, int N, int K, const float* PW, const float* b,
                  float* Y, int mode, const float* Xin, const float* skip) {
    hgt_gemm_wmma<<<dim3((N + 127) / 128), dim3(256), 0, stream>>>(
        X, PW, b, Y, Xin, skip, N, K, mode);
  };
  auto packw = [&](const float* W, float* PW, int K) {
    hgt_pack_w<<<(K * HID + 255) / 256, 256, 0, stream>>>(W, PW, K);
  };

  // input projection + relu (pack then GEMM)
  for (int t = 0; t < 3; ++t) {
    int wi = 11 + pslot[t] * 2;
    packw(P(wi), packIn + (size_t)t * 16384, Kin[t]);
    gemm(x_in[t], Nt[t], Kin[t], packIn + (size_t)t * 16384, P(wi + 1),
         bufA + base[t] * HID, 1, nullptr, nullptr);
  }

  float* xcur = bufA;
  for (int l = 0; l < 2; ++l) {
    const int L = 17 + l * 39;
    const int Ga = L, Gk = L + 6, Gq = L + 12, Gr = L + 18, Gs = L + 30, Gv = L + 33;
    float* xout = (l == 0) ? bufB : (float*)d_out;

    // fold a_rel into K-weights, m_rel into V-weights (packed output)
    const int fblk = (128 * 128 + 128 + 255) / 256;
    for (int r = 0; r < 4; ++r) {
      int s = rsrc[r];
      int kw = Gk + pslot[s] * 2, vw = Gv + pslot[s] * 2, rp = Gr + rslot[r] * 3;
      hgt_fold_rel<<<fblk, 256, 0, stream>>>(P(kw), P(kw + 1), P(rp + 0),
          foldW + (size_t)(r * 2) * 16384, foldB + (r * 2) * 128);
      hgt_fold_rel<<<fblk, 256, 0, stream>>>(P(vw), P(vw + 1), P(rp + 1),
          foldW + (size_t)(r * 2 + 1) * 16384, foldB + (r * 2 + 1) * 128);
    }
    // pack q/a weights for this layer
    for (int t = 0; t < 3; ++t) {
      packw(P(Gq + pslot[t] * 2), packQ + (size_t)t * 16384, HID);
      packw(P(Ga + pslot[t] * 2), packA + (size_t)t * 16384, HID);
    }
    // q per type
    for (int t = 0; t < 3; ++t) {
      gemm(xcur + base[t] * HID, Nt[t], HID, packQ + (size_t)t * 16384,
           P(Gq + pslot[t] * 2 + 1), qbuf + base[t] * HID, 0, nullptr, nullptr);
    }
    // k_rel / msg per relation (folded packed weights)
    for (int r = 0; r < 4; ++r) {
      int s = rsrc[r];
      gemm(xcur + base[s] * HID, Nt[s], HID, foldW + (size_t)(r * 2) * 16384,
           foldB + (r * 2) * 128, krbuf + relbase[r] * HID, 0, nullptr, nullptr);
      gemm(xcur + base[s] * HID, Nt[s], HID, foldW + (size_t)(r * 2 + 1) * 16384,
           foldB + (r * 2 + 1) * 128, mgbuf + relbase[r] * HID, 0, nullptr, nullptr);
    }
    // reset segment-softmax state
    hipMemsetAsync(mmax,   0, (size_t)NT * 8 * sizeof(unsigned), stream);
    hipMemsetAsync(den,    0, (size_t)NT * 8 * sizeof(float), stream);
    hipMemsetAsync(aggbuf, 0, (size_t)NT * HID * sizeof(float), stream);
    // pass 1: scores + max
    for (int r = 0; r < 4; ++r) {
      int rp = Gr + rslot[r] * 3;
      hgt_edge_score<<<(NE * 8 + 255) / 256, 256, 0, stream>>>(
          ei[r], ei[r] + NE, qbuf + base[rdst[r]] * HID, krbuf + relbase[r] * HID,
          P(rp + 2), scoreb + (size_t)r * NE * 8, mmax + base[rdst[r]] * 8, NE);
    }
    // pass 2: exp, denominator, aggregate
    for (int r = 0; r < 4; ++r) {
      hgt_edge_accum<<<(NE * 8 + 255) / 256, 256, 0, stream>>>(
          ei[r], ei[r] + NE, ew[r], scoreb + (size_t)r * NE * 8,
          mgbuf + relbase[r] * HID, mmax + base[rdst[r]] * 8,
          den + base[rdst[r]] * 8, aggbuf + base[rdst[r]] * HID, NE);
    }
    // normalize + gelu
    hgt_norm_gelu<<<((size_t)NT * HID + 255) / 256, 256, 0, stream>>>(aggbuf, den, NT);
    // output projection + skip gate
    for (int t = 0; t < 3; ++t) {
      gemm(aggbuf + base[t] * HID, Nt[t], HID, packA + (size_t)t * 16384,
           P(Ga + pslot[t] * 2 + 1), xout + base[t] * HID, 2,
           xcur + base[t] * HID, P(Gs + pslot[t]));
    }
    xcur = xout;
  }
}